// PKNeuralODE_56530359550046
// MI455X (gfx1250) — compile-verified
//
#include <hip/hip_runtime.h>
#include <math.h>

typedef __attribute__((ext_vector_type(2))) float v2f;
typedef __attribute__((ext_vector_type(8))) float v8f;

namespace {

constexpr int kHidden = 64;
constexpr int kNT     = 385;
constexpr int kNDose  = 8;
constexpr int kSub    = 16;

// Fast SiLU: x * sigmoid(x) with v_exp_f32 + v_rcp_f32 (avoids the ~8-op IEEE
// divide sequence on the serial critical path; error << dopri5 truncation).
__device__ __forceinline__ float silu_f(float x) {
    return x * __builtin_amdgcn_rcpf(1.0f + __expf(-x));
}

// out = silu(W(64x64, row-major in LDS) @ hin(64) + b), via V_WMMA_F32_16X16X4_F32.
// One wave32. The h-vector is replicated across all 16 B-columns, so after the
// K accumulation every column of the 16x16 accumulator holds the full matvec
// for its row -> no cross-lane reduction. Lanes 0 and 16 own rows
// [16r+0..7] / [16r+8..15] of the D fragment (ISA 16x16 f32 C/D layout).
// Two interleaved accumulators halve the dependent WMMA chain depth (16 -> 8).
__device__ __forceinline__ void matvec64(const float* __restrict__ Wlds,
                                         const float* __restrict__ blds,
                                         const float* __restrict__ hin,
                                         float* __restrict__ hout,
                                         int lane) {
    const int half = lane >> 4;   // 0 -> K pair {0,1}, 1 -> K pair {2,3}
    const int m15  = lane & 15;   // row within tile (A), column (B/D)
#pragma unroll
    for (int r = 0; r < 4; ++r) {
        v8f acc0 = {};
        v8f acc1 = {};
#pragma unroll
        for (int kk = 0; kk < 16; kk += 2) {
            const int kb0 = 4 * kk + 2 * half;
            const int kb1 = kb0 + 4;
            // A 16x4 f32 fragment: lane<16 holds A[m,{0,1}], lane>=16 holds A[m,{2,3}]
            v2f a0 = *(const v2f*)(Wlds + (16 * r + m15) * kHidden + kb0);
            v2f a1 = *(const v2f*)(Wlds + (16 * r + m15) * kHidden + kb1);
            // B 4x16 f32 fragment, identical in every column n
            v2f b0 = *(const v2f*)(hin + kb0);
            v2f b1 = *(const v2f*)(hin + kb1);
            acc0 = __builtin_amdgcn_wmma_f32_16x16x4_f32(false, a0, false, b0,
                                                         (short)0, acc0, false, false);
            acc1 = __builtin_amdgcn_wmma_f32_16x16x4_f32(false, a1, false, b1,
                                                         (short)0, acc1, false, false);
        }
        v8f acc = acc0 + acc1;
        if (m15 == 0) {           // lanes 0 and 16 write their 8 rows
            const int base = 16 * r + 8 * half;
#pragma unroll
            for (int v = 0; v < 8; ++v) {
                float x = acc[v] + blds[base + v];
                hout[base + v] = silu_f(x);
            }
        }
    }
    __syncthreads();
}

// dy/dt = MLP(y): 1 -> 64 -> 64 -> 64 -> 1, SiLU on hidden layers.
__device__ __forceinline__ float f_eval(float y,
                                        const float* __restrict__ sW1,
                                        const float* __restrict__ sb1,
                                        const float* __restrict__ sW2,
                                        const float* __restrict__ sb2,
                                        const float* __restrict__ sW3,
                                        const float* __restrict__ sb3,
                                        const float* __restrict__ sW4,
                                        float b4,
                                        float* __restrict__ h1,
                                        float* __restrict__ h2,
                                        float* __restrict__ h3,
                                        int lane) {
    // Layer 1: elementwise (input is scalar y); each lane produces 2 elements.
    h1[lane]      = silu_f(sW1[lane]      * y + sb1[lane]);
    h1[lane + 32] = silu_f(sW1[lane + 32] * y + sb1[lane + 32]);
    __syncthreads();
    // Layers 2 and 3: 64x64 matvecs on the WMMA pipe (f32 precision).
    matvec64(sW2, sb2, h1, h2, lane);
    matvec64(sW3, sb3, h2, h3, lane);
    // Layer 4: 64-length dot product, wave32 butterfly reduction.
    float p = sW4[lane] * h3[lane] + sW4[lane + 32] * h3[lane + 32];
#pragma unroll
    for (int m = 16; m >= 1; m >>= 1) p += __shfl_xor(p, m, 32);
    return p + b4;
}

__global__ __launch_bounds__(32, 1)
void pk_node_kernel(const float* __restrict__ gTimes,
                    const float* __restrict__ gDoseT,
                    const float* __restrict__ gDoseA,
                    const float* __restrict__ gLogDE,
                    const float* __restrict__ gW1, const float* __restrict__ gb1,
                    const float* __restrict__ gW2, const float* __restrict__ gb2,
                    const float* __restrict__ gW3, const float* __restrict__ gb3,
                    const float* __restrict__ gW4, const float* __restrict__ gb4,
                    float* __restrict__ out) {
    __shared__ float sW2[kHidden * kHidden];
    __shared__ float sW3[kHidden * kHidden];
    __shared__ float sW1[kHidden], sb1[kHidden], sb2[kHidden], sb3[kHidden], sW4[kHidden];
    __shared__ float sTimes[kNT];
    __shared__ float sDT[kNDose], sDA[kNDose];
    __shared__ float h1[kHidden], h2[kHidden], h3[kHidden];

    const int lane = threadIdx.x;

    // Stage all parameters into LDS once (~36 KB of 320 KB).
    for (int i = lane; i < kHidden * kHidden; i += 32) { sW2[i] = gW2[i]; sW3[i] = gW3[i]; }
    for (int i = lane; i < kHidden; i += 32) {
        sW1[i] = gW1[i]; sb1[i] = gb1[i]; sb2[i] = gb2[i]; sb3[i] = gb3[i]; sW4[i] = gW4[i];
    }
    for (int i = lane; i < kNT; i += 32) sTimes[i] = gTimes[i];
    if (lane < kNDose) { sDT[lane] = gDoseT[lane]; sDA[lane] = gDoseA[lane]; }
    const float b4      = gb4[0];
    const float doseEff = __expf(gLogDE[0]);
    __syncthreads();

    // Dormand-Prince 5(4) tableau (5th-order weights; stage 7 weight is zero).
    const float cA[6][5] = {
        {0.f, 0.f, 0.f, 0.f, 0.f},
        {(float)(1.0/5.0), 0.f, 0.f, 0.f, 0.f},
        {(float)(3.0/40.0), (float)(9.0/40.0), 0.f, 0.f, 0.f},
        {(float)(44.0/45.0), (float)(-56.0/15.0), (float)(32.0/9.0), 0.f, 0.f},
        {(float)(19372.0/6561.0), (float)(-25360.0/2187.0), (float)(64448.0/6561.0),
         (float)(-212.0/729.0), 0.f},
        {(float)(9017.0/3168.0), (float)(-355.0/33.0), (float)(46732.0/5247.0),
         (float)(49.0/176.0), (float)(-5103.0/18656.0)}};
    const float cB1 = (float)(35.0/384.0),   cB3 = (float)(500.0/1113.0);
    const float cB4 = (float)(125.0/192.0),  cB5 = (float)(-2187.0/6784.0);
    const float cB6 = (float)(11.0/84.0);

    float y = 0.0f;
#pragma unroll 1
    for (int i = 0; i < kNT; ++i) {
        const float t  = sTimes[i];
        const float dt = (i + 1 < kNT) ? (sTimes[i + 1] - t) : 0.0f;

        // Dose event: fires when prod(t - dose_times)/1e7 ~ 0.
        float ev = 1.0f, dose = 0.0f;
#pragma unroll
        for (int j = 0; j < kNDose; ++j) {
            const float d = t - sDT[j];
            ev *= d;
            if (fabsf(d) < 1e-6f) dose += sDA[j];
        }
        ev *= 1e-7f;  // / 10^(n_dose-1)
        if (fabsf(ev) < 1e-6f) y += dose * doseEff;

        if (lane == 0) out[i] = y;  // pred recorded after dose, before integration

        const float h = dt * (1.0f / kSub);
#pragma unroll 1
        for (int s = 0; s < kSub; ++s) {
            float ks[6];
#pragma unroll 1
            for (int st = 0; st < 6; ++st) {
                float yy = y;
                for (int j = 0; j < st; ++j) yy += h * cA[st][j] * ks[j];
                ks[st] = f_eval(yy, sW1, sb1, sW2, sb2, sW3, sb3, sW4, b4,
                                h1, h2, h3, lane);
            }
            y += h * (cB1 * ks[0] + cB3 * ks[2] + cB4 * ks[3] + cB5 * ks[4] + cB6 * ks[5]);
        }
    }
}

}  // namespace

extern "C" void kernel_launch(void* const* d_in, const int* in_sizes, int n_in,
                              void* d_out, int out_size, void* d_ws, size_t ws_size,
                              hipStream_t stream) {
    (void)in_sizes; (void)n_in; (void)out_size; (void)d_ws; (void)ws_size;
    const float* times = (const float*)d_in[0];
    const float* doseT = (const float*)d_in[1];
    const float* doseA = (const float*)d_in[2];
    const float* logDE = (const float*)d_in[3];
    const float* W1 = (const float*)d_in[4];
    const float* b1 = (const float*)d_in[5];
    const float* W2 = (const float*)d_in[6];
    const float* b2 = (const float*)d_in[7];
    const float* W3 = (const float*)d_in[8];
    const float* b3 = (const float*)d_in[9];
    const float* W4 = (const float*)d_in[10];
    const float* b4 = (const float*)d_in[11];

    // Strictly sequential scan -> one persistent wave32 workgroup.
    pk_node_kernel<<<1, 32, 0, stream>>>(times, doseT, doseA, logDE,
                                         W1, b1, W2, b2, W3, b3, W4, b4,
                                         (float*)d_out);
}